// Sinkhorn_1829656068572
// MI455X (gfx1250) — compile-verified
//
#include <hip/hip_runtime.h>
#include <math.h>

#define EPSV     0.1f
#define INV_EPS  10.0f
#define NN       2048
#define DD       64
#define BB       4
#define MAX_ITER 100
#define THRESHV  1e-3f
#define BIG_NEG  -1e30f

typedef __attribute__((ext_vector_type(2))) float v2f;
typedef __attribute__((ext_vector_type(8))) float v8f;

// workspace layout (float offsets)
enum {
  OFF_X2   = 0,        // B*N
  OFF_Y2   = 8192,
  OFF_U    = 16384,
  OFF_V    = 24576,
  OFF_UT   = 32768,
  OFF_VT   = 40960,
  OFF_CSUM = 49152,    // 4 chunks * B * N = 32768
  OFF_BMAX = 81920,    // 2048 gemm-block maxes
  OFF_BCST = 83968,    // 4096 pi-block cost partials
  OFF_SCAL = 88064     // [0]=invMax, [1]=done
};

// ---- squared row norms of x and y -------------------------------------------------
__global__ void sq_norm_kernel(const float* __restrict__ x, const float* __restrict__ y,
                               float* __restrict__ x2, float* __restrict__ y2) {
  int t = blockIdx.x * 256 + threadIdx.x;          // 0..16383
  int isY = (t >= BB * NN);
  int r = isY ? (t - BB * NN) : t;
  const float4* p = (const float4*)((isY ? y : x) + (size_t)r * DD);
  float s = 0.0f;
#pragma unroll
  for (int i = 0; i < DD / 4; ++i) {
    float4 a = p[i];
    s += a.x * a.x + a.y * a.y + a.z * a.z + a.w * a.w;
  }
  if (isY) y2[r] = s; else x2[r] = s;
}

// ---- init u, v, done --------------------------------------------------------------
__global__ void init_kernel(float* __restrict__ ws) {
  int t = blockIdx.x * 256 + threadIdx.x;          // 0..8191
  ws[OFF_U + t] = 0.0f;
  ws[OFF_V + t] = 0.0f;
  if (t == 0) ws[OFF_SCAL + 1] = 0.0f;             // done = false
}

// ---- cost-matrix GEMM via V_WMMA_F32_16X16X4_F32 ----------------------------------
// grid: B * (N/16) * (M/512) = 2048 blocks, 256 threads.
// Each wave computes a 16x64 tile (4 accumulators) so the A fragment is loaded once
// per k-step and reused 4x: 5 global_load_b64 per 4 WMMAs.
__global__ void gemm_cost_kernel(const float* __restrict__ x, const float* __restrict__ y,
                                 const float* __restrict__ ws, float* __restrict__ C,
                                 float* __restrict__ blockMax) {
  int bid   = blockIdx.x;
  int b     = bid >> 9;                            // / (128*4)
  int rem   = bid & 511;
  int ntile = rem >> 2;                            // 0..127
  int mblk  = rem & 3;                             // 0..3
  int wave  = threadIdx.x >> 5;
  int lane  = threadIdx.x & 31;
  int lm    = lane & 15;
  int kb    = (lane >> 4) << 1;                    // lanes 0-15: K base 0, lanes 16-31: K base 2
  int n0    = ntile << 4;
  int m00   = (mblk << 9) + (wave << 6);           // wave's 64-wide m range

  const float* xrow = x + ((size_t)(b * NN + n0 + lm)) * DD + kb;   // A: 16x4 fragment rows
  const float* yrow = y + ((size_t)(b * NN + m00 + lm)) * DD + kb;  // B = y^T, subtile j at +j*16 rows

  v8f acc0 = {}, acc1 = {}, acc2 = {}, acc3 = {};
#pragma unroll
  for (int k = 0; k < DD; k += 4) {
    v2f a  = *(const v2f*)(xrow + k);
    v2f b0 = *(const v2f*)(yrow + k);
    v2f b1 = *(const v2f*)(yrow + 16 * DD + k);
    v2f b2 = *(const v2f*)(yrow + 32 * DD + k);
    v2f b3 = *(const v2f*)(yrow + 48 * DD + k);
    acc0 = __builtin_amdgcn_wmma_f32_16x16x4_f32(false, a, false, b0, (short)0, acc0, false, false);
    acc1 = __builtin_amdgcn_wmma_f32_16x16x4_f32(false, a, false, b1, (short)0, acc1, false, false);
    acc2 = __builtin_amdgcn_wmma_f32_16x16x4_f32(false, a, false, b2, (short)0, acc2, false, false);
    acc3 = __builtin_amdgcn_wmma_f32_16x16x4_f32(false, a, false, b3, (short)0, acc3, false, false);
  }

  const float* x2 = ws + OFF_X2;
  const float* y2 = ws + OFF_Y2;
  int rowAdd = (lane >> 4) << 3;                   // upper half-lanes hold M = r + 8
  float x2v[8];
#pragma unroll
  for (int r = 0; r < 8; ++r) x2v[r] = x2[b * NN + n0 + r + rowAdd];

  float lmax = BIG_NEG;
#pragma unroll
  for (int j = 0; j < 4; ++j) {
    v8f acc = (j == 0) ? acc0 : (j == 1) ? acc1 : (j == 2) ? acc2 : acc3;
    int col = m00 + j * 16 + lm;
    float y2v = y2[b * NN + col];
#pragma unroll
    for (int r = 0; r < 8; ++r) {
      int row = n0 + r + rowAdd;
      float val = x2v[r] + y2v - 2.0f * acc[r];
      C[((size_t)(b * NN + row)) * NN + col] = val;
      lmax = fmaxf(lmax, val);
    }
  }

  __shared__ float smax[256];
  smax[threadIdx.x] = lmax;
  __syncthreads();
  for (int s = 128; s > 0; s >>= 1) {
    if (threadIdx.x < s) smax[threadIdx.x] = fmaxf(smax[threadIdx.x], smax[threadIdx.x + s]);
    __syncthreads();
  }
  if (threadIdx.x == 0) blockMax[bid] = smax[0];
}

// ---- global max -> 1/max ----------------------------------------------------------
__global__ void reduce_max_kernel(const float* __restrict__ blockMax, float* __restrict__ scal) {
  float m = BIG_NEG;
  for (int i = threadIdx.x; i < 2048; i += 256) m = fmaxf(m, blockMax[i]);
  __shared__ float sm[256];
  sm[threadIdx.x] = m;
  __syncthreads();
  for (int s = 128; s > 0; s >>= 1) {
    if (threadIdx.x < s) sm[threadIdx.x] = fmaxf(sm[threadIdx.x], sm[threadIdx.x + s]);
    __syncthreads();
  }
  if (threadIdx.x == 0) scal[0] = 1.0f / sm[0];
}

// ---- normalize C in place ---------------------------------------------------------
__global__ void normalize_kernel(float* __restrict__ C, const float* __restrict__ scal) {
  float s = scal[0];
  float4* P = (float4*)C;
  size_t base = (size_t)blockIdx.x * 1024;
#pragma unroll
  for (int j = 0; j < 4; ++j) {
    size_t i = base + j * 256 + threadIdx.x;
    float4 c = P[i];
    c.x *= s; c.y *= s; c.z *= s; c.w *= s;
    P[i] = c;
  }
}

// ---- row pass: u_tmp[b][n] = eps*(log_mu - log sum_m exp((v[m]-C[n][m])/eps)) ------
// Plain exp-sum is numerically safe here: after normalization C in [0,1] and u,v stay
// O(1), so the argument lies in ~[-25, +12], far inside f32 exp range. One wave/row.
__global__ void row_lse_kernel(const float* __restrict__ C, const float* __restrict__ ws_v,
                               float* __restrict__ u_tmp, float log_mu) {
  int wave = threadIdx.x >> 5;
  int lane = threadIdx.x & 31;
  int r = blockIdx.x * 8 + wave;                   // r = b*N + n
  int b = r >> 11;
  const float4* Crow = (const float4*)(C + (size_t)r * NN);
  const float4* vrow = (const float4*)(ws_v + b * NN);
  float sm = 0.0f;
#pragma unroll 4
  for (int i = 0; i < 16; ++i) {
    int idx = i * 32 + lane;
    float4 c4 = Crow[idx];
    float4 v4 = vrow[idx];
    sm += __expf((v4.x - c4.x) * INV_EPS);
    sm += __expf((v4.y - c4.y) * INV_EPS);
    sm += __expf((v4.z - c4.z) * INV_EPS);
    sm += __expf((v4.w - c4.w) * INV_EPS);
  }
#pragma unroll
  for (int off = 16; off > 0; off >>= 1) sm += __shfl_xor(sm, off, 32);
  if (lane == 0) u_tmp[r] = EPSV * (log_mu - __logf(sm));
}

// ---- column pass partials: sum_n exp((u_new[n]-C[n][m])/eps) over 4 n-chunks ------
// grid: B * 8 mtiles * 4 chunks = 128 blocks, 256 threads (thread = one column)
__global__ void col_lse_partial_kernel(const float* __restrict__ C, const float* __restrict__ u_tmp,
                                       float* __restrict__ csum) {
  int bid   = blockIdx.x;
  int chunk = bid & 3;
  int t2    = bid >> 2;
  int mtile = t2 & 7;
  int b     = t2 >> 3;
  int m     = mtile * 256 + threadIdx.x;
  int n0    = chunk * 512;
  const float* Cb = C + (size_t)b * NN * NN;
  const float* ub = u_tmp + b * NN;
  float sm = 0.0f;
#pragma unroll 4
  for (int n = n0; n < n0 + 512; ++n) {
    sm += __expf((ub[n] - Cb[(size_t)n * NN + m]) * INV_EPS);
  }
  csum[(chunk * BB + b) * NN + m] = sm;
}

// ---- combine chunk partials -> v_tmp ----------------------------------------------
__global__ void col_combine_kernel(const float* __restrict__ csum,
                                   float* __restrict__ v_tmp, float log_nu) {
  int i = blockIdx.x * 256 + threadIdx.x;          // i = b*N + m
  int b = i >> 11;
  int m = i & 2047;
  float sm = 0.0f;
#pragma unroll
  for (int c = 0; c < 4; ++c) sm += csum[(c * BB + b) * NN + m];
  v_tmp[i] = EPSV * (log_nu - __logf(sm));
}

// ---- commit with done-gating + convergence check (single block) -------------------
__global__ void commit_kernel(float* __restrict__ ws) {
  float* u        = ws + OFF_U;
  float* v        = ws + OFF_V;
  const float* ut = ws + OFF_UT;
  const float* vt = ws + OFF_VT;
  float done = ws[OFF_SCAL + 1];                   // pre-iteration flag (gates updates)
  int t = threadIdx.x;
  float errloc = 0.0f;
  for (int i = t; i < BB * NN; i += 1024) {
    float un = ut[i];
    errloc += fabsf(un - u[i]);
    if (done == 0.0f) { u[i] = un; v[i] = vt[i]; }
  }
  __shared__ float se[1024];
  se[t] = errloc;
  __syncthreads();
  for (int s = 512; s > 0; s >>= 1) {
    if (t < s) se[t] += se[t + s];
    __syncthreads();
  }
  if (t == 0) {
    float err = se[0] * (1.0f / BB);               // mean over batches of sum |du|
    if (err < THRESHV) ws[OFF_SCAL + 1] = 1.0f;    // sticky
  }
}

// ---- pi = exp((-C+u+v)/eps), partial cost sums ------------------------------------
// grid: 4096 blocks (1024 per batch), 256 threads, 4096 elems / block
__global__ void pi_cost_kernel(const float* __restrict__ C, const float* __restrict__ ws,
                               float* __restrict__ pi, float* __restrict__ blockCost) {
  int bid = blockIdx.x;
  int b   = bid >> 10;
  int blk = bid & 1023;
  size_t base = (size_t)b * NN * NN + (size_t)blk * 4096;
  const float4* Cp = (const float4*)(C + base);
  float4* Pp = (float4*)(pi + base);
  const float* ub = ws + OFF_U + b * NN;
  const float* vb = ws + OFF_V + b * NN;
  float acc = 0.0f;
#pragma unroll
  for (int j = 0; j < 4; ++j) {
    int i4 = j * 256 + threadIdx.x;
    int e  = blk * 4096 + i4 * 4;
    int n  = e >> 11;
    int m  = e & 2047;
    float4 c4 = Cp[i4];
    float uv = ub[n];
    float4 p4;
    p4.x = __expf((uv + vb[m + 0] - c4.x) * INV_EPS);
    p4.y = __expf((uv + vb[m + 1] - c4.y) * INV_EPS);
    p4.z = __expf((uv + vb[m + 2] - c4.z) * INV_EPS);
    p4.w = __expf((uv + vb[m + 3] - c4.w) * INV_EPS);
    Pp[i4] = p4;
    acc += p4.x * c4.x + p4.y * c4.y + p4.z * c4.z + p4.w * c4.w;
  }
  __shared__ float sa[256];
  sa[threadIdx.x] = acc;
  __syncthreads();
  for (int s = 128; s > 0; s >>= 1) {
    if (threadIdx.x < s) sa[threadIdx.x] += sa[threadIdx.x + s];
    __syncthreads();
  }
  if (threadIdx.x == 0) blockCost[bid] = sa[0];
}

// ---- final per-batch cost ---------------------------------------------------------
__global__ void cost_reduce_kernel(const float* __restrict__ blockCost, float* __restrict__ cost) {
  int b = blockIdx.x;
  float s = 0.0f;
  for (int i = threadIdx.x; i < 1024; i += 256) s += blockCost[b * 1024 + i];
  __shared__ float sa[256];
  sa[threadIdx.x] = s;
  __syncthreads();
  for (int st = 128; st > 0; st >>= 1) {
    if (threadIdx.x < st) sa[threadIdx.x] += sa[threadIdx.x + st];
    __syncthreads();
  }
  if (threadIdx.x == 0) cost[b] = sa[0];
}

extern "C" void kernel_launch(void* const* d_in, const int* in_sizes, int n_in,
                              void* d_out, int out_size, void* d_ws, size_t ws_size,
                              hipStream_t stream) {
  const float* x = (const float*)d_in[0];
  const float* y = (const float*)d_in[1];
  float* out  = (float*)d_out;
  float* cost = out;                               // [4]
  float* pi   = out + 4;                           // [4,2048,2048]
  float* Cmat = out + 4 + (size_t)BB * NN * NN;    // [4,2048,2048], normalized
  float* ws   = (float*)d_ws;

  const float log_mu = logf(1.0f / (float)NN + 1e-8f);  // == log_nu (N==M)

  // 1) squared norms of x rows and y rows
  sq_norm_kernel<<<64, 256, 0, stream>>>(x, y, ws + OFF_X2, ws + OFF_Y2);
  // 2) u=v=0, done=0
  init_kernel<<<32, 256, 0, stream>>>(ws);
  // 3) raw cost matrix via f32 WMMA (16x64 tile / wave, A-fragment reuse) + block maxes
  gemm_cost_kernel<<<2048, 256, 0, stream>>>(x, y, ws, Cmat, ws + OFF_BMAX);
  // 4) global max -> invMax
  reduce_max_kernel<<<1, 256, 0, stream>>>(ws + OFF_BMAX, ws + OFF_SCAL);
  // 5) C *= invMax (in place; stays L2-resident for the iteration loop)
  normalize_kernel<<<4096, 256, 0, stream>>>(Cmat, ws + OFF_SCAL);

  // 6) Sinkhorn iterations (fixed trip count, device-side done gating like the reference)
  for (int it = 0; it < MAX_ITER; ++it) {
    row_lse_kernel<<<1024, 256, 0, stream>>>(Cmat, ws + OFF_V, ws + OFF_UT, log_mu);
    col_lse_partial_kernel<<<128, 256, 0, stream>>>(Cmat, ws + OFF_UT, ws + OFF_CSUM);
    col_combine_kernel<<<32, 256, 0, stream>>>(ws + OFF_CSUM, ws + OFF_VT, log_mu);
    commit_kernel<<<1, 1024, 0, stream>>>(ws);
  }

  // 7) transport plan + partial costs
  pi_cost_kernel<<<4096, 256, 0, stream>>>(Cmat, ws, pi, ws + OFF_BCST);
  // 8) per-batch cost
  cost_reduce_kernel<<<BB, 256, 0, stream>>>(ws + OFF_BCST, cost);
}